// ScreenShooting_10230612099394
// MI455X (gfx1250) — compile-verified
//
#include <hip/hip_runtime.h>
#include <math.h>

#define HD 256
#define WD 256
#define BD 64
#define HW (HD * WD)

typedef float v2f __attribute__((ext_vector_type(2)));
typedef float v8f __attribute__((ext_vector_type(8)));

// ---------------------------------------------------------------------------
// Per-batch homography: solve 8x8 system (partial-pivot GE), then invert the
// 3x3 homography.
// ---------------------------------------------------------------------------
__device__ __forceinline__ void compute_hinv(const float* __restrict__ off, float* out9) {
    const float sx[4] = {0.f, 255.f, 255.f, 0.f};
    const float sy[4] = {0.f, 0.f, 255.f, 255.f};
    const float dx[4] = {0.f, 256.f, 256.f, 0.f};
    const float dy[4] = {0.f, 0.f, 256.f, 256.f};
    float M[8][9];
    for (int i = 0; i < 4; ++i) {
        float u = dx[i] + off[2 * i + 0];
        float v = dy[i] + off[2 * i + 1];
        float x = sx[i], y = sy[i];
        float* r0 = M[i];
        float* r1 = M[i + 4];
        r0[0] = x;   r0[1] = y;   r0[2] = 1.f; r0[3] = 0.f; r0[4] = 0.f; r0[5] = 0.f;
        r0[6] = -u * x; r0[7] = -u * y; r0[8] = u;
        r1[0] = 0.f; r1[1] = 0.f; r1[2] = 0.f; r1[3] = x;   r1[4] = y;   r1[5] = 1.f;
        r1[6] = -v * x; r1[7] = -v * y; r1[8] = v;
    }
    for (int k = 0; k < 8; ++k) {
        int p = k; float best = fabsf(M[k][k]);
        for (int r = k + 1; r < 8; ++r) { float a = fabsf(M[r][k]); if (a > best) { best = a; p = r; } }
        if (p != k) for (int c = k; c < 9; ++c) { float t = M[k][c]; M[k][c] = M[p][c]; M[p][c] = t; }
        float inv = 1.f / M[k][k];
        for (int r = k + 1; r < 8; ++r) {
            float f = M[r][k] * inv;
            for (int c = k; c < 9; ++c) M[r][c] -= f * M[k][c];
        }
    }
    float h[9];
    for (int k = 7; k >= 0; --k) {
        float s = M[k][8];
        for (int c = k + 1; c < 8; ++c) s -= M[k][c] * h[c];
        h[k] = s / M[k][k];
    }
    h[8] = 1.f;
    float a = h[0], b = h[1], c = h[2], d = h[3], e = h[4], f = h[5], g = h[6], hh = h[7], i = h[8];
    float A0 = e * i - f * hh;
    float A1 = -(d * i - f * g);
    float A2 = d * hh - e * g;
    float det = a * A0 + b * A1 + c * A2;
    float id = 1.f / det;
    out9[0] = A0 * id;                 out9[1] = -(b * i - c * hh) * id;  out9[2] = (b * f - c * e) * id;
    out9[3] = A1 * id;                 out9[4] = (a * i - c * g) * id;    out9[5] = -(a * f - c * d) * id;
    out9[6] = A2 * id;                 out9[7] = -(a * hh - b * g) * id;  out9[8] = (a * e - b * d) * id;
}

__device__ __forceinline__ float moire2(int row, int col, int c,
                                        const float* __restrict__ moire_c,
                                        const int* __restrict__ theta) {
    float fi = (float)row + 1.f;
    float fj = (float)col + 1.f;
    float cx = moire_c[2 * c + 0];
    float cy = moire_c[2 * c + 1];
    float th = (float)theta[c] / 180.f * 3.14159265358979323846f;
    float dx = fi - cx, dy = fj - cy;
    float z1 = 0.5f + 0.5f * cosf(6.2831853071795864769f * sqrtf(dx * dx + dy * dy));
    float z2 = 0.5f + 0.5f * cosf(cosf(th) * fj + sinf(th) * fi);
    float mo = (fminf(z1, z2) + 1.f) * 0.5f;
    return mo * 2.f - 1.f;
}

// One thread per batch: Hinv -> ws (runs once, off the hot path).
__global__ void hinv_prep_kernel(const float* __restrict__ offs,
                                 float* __restrict__ hinv_out) {
    int b = blockIdx.x * blockDim.x + threadIdx.x;
    if (b < BD) compute_hinv(offs + b * 8, hinv_out + b * 9);
}

// Precompute moire (batch-invariant, 3*H*W) once into L2-resident scratch.
__global__ void moire_prep_kernel(const float* __restrict__ moire_c,
                                  const int* __restrict__ theta,
                                  float* __restrict__ mo_out) {
    int idx = blockIdx.x * blockDim.x + threadIdx.x;
    if (idx >= 3 * HW) return;
    int c = idx / HW;
    int rem = idx - c * HW;
    int row = rem >> 8;
    int col = rem & 255;
    mo_out[idx] = moire2(row, col, c, moire_c, theta);
}

// ---------------------------------------------------------------------------
// Main fused kernel. Block = one (batch,row); thread = one column; all 3
// channels per thread. Projective coords p = Hinv @ [x,y,1]^T computed with
// V_WMMA_F32_16X16X4_F32 (two 16-pixel halves per wave32), matching the
// reference einsum('rc,chw->rhw').
// ---------------------------------------------------------------------------
template <bool PRE>
__global__ void __launch_bounds__(256)
shoot_kernel(const float* __restrict__ img,
             const float* __restrict__ offs,
             const float* __restrict__ ab,
             const float* __restrict__ moire_c,
             const float* __restrict__ noise,
             const int* __restrict__ light_xy,
             const int* __restrict__ theta,
             const float* __restrict__ mo_pre,
             const float* __restrict__ hinv_pre,
             float* __restrict__ out) {
    int blk = blockIdx.x;        // b * H + row
    int b = blk >> 8;
    int row = blk & 255;

    int col = threadIdx.x;       // 0..255
    int lid = col & 31;          // wave32 lane
    int waveBase = col & ~31;
    int m = lid & 15;
    bool hi = lid >= 16;         // hi lanes carry K2/K3

    // --- fetch/compute Hinv (uniform per block) ---
    float h0, h1, h2, h3, h4, h5, h6, h7, h8;
    if (PRE) {
        const float* hv = hinv_pre + b * 9;   // uniform -> scalar loads
        h0 = hv[0]; h1 = hv[1]; h2 = hv[2];
        h3 = hv[3]; h4 = hv[4]; h5 = hv[5];
        h6 = hv[6]; h7 = hv[7]; h8 = hv[8];
    } else {
        __shared__ float sH[9];
        if (threadIdx.x == 0) compute_hinv(offs + b * 8, sH);
        __syncthreads();
        h0 = sH[0]; h1 = sH[1]; h2 = sH[2];
        h3 = sH[3]; h4 = sH[4]; h5 = sH[5];
        h6 = sH[6]; h7 = sH[7]; h8 = sH[8];
    }

    // --- WMMA fragments (ISA 16x4 f32 A layout; 4x16 B layout) ---
    float colK0 = (m == 0) ? h0 : (m == 1) ? h3 : (m == 2) ? h6 : 0.f;
    float colK1 = (m == 0) ? h1 : (m == 1) ? h4 : (m == 2) ? h7 : 0.f;
    float colK2 = (m == 0) ? h2 : (m == 1) ? h5 : (m == 2) ? h8 : 0.f;
    v2f afrag;
    afrag.x = hi ? colK2 : colK0;   // VGPR0: K0 (lo lanes) | K2 (hi lanes)
    afrag.y = hi ? 0.f   : colK1;   // VGPR1: K1 (lo lanes) | K3=0 (hi lanes)

    float fy = (float)row;
    float xn0 = (float)(waveBase + m);
    float xn1 = xn0 + 16.f;
    v2f b0, b1;
    b0.x = hi ? 1.f : xn0;  b0.y = hi ? 0.f : fy;   // B rows: x, y, 1, 0
    b1.x = hi ? 1.f : xn1;  b1.y = hi ? 0.f : fy;

    v8f cz = {0.f, 0.f, 0.f, 0.f, 0.f, 0.f, 0.f, 0.f};
    v8f d0 = __builtin_amdgcn_wmma_f32_16x16x4_f32(false, afrag, false, b0,
                                                   (short)0, cz, false, false);
    v8f d1 = __builtin_amdgcn_wmma_f32_16x16x4_f32(false, afrag, false, b1,
                                                   (short)0, cz, false, false);

    // Rows 0..2 of D (lanes 0-15) hold p0,p1,p2; shuffle second half up.
    float q0 = __shfl(d1[0], m, 32);
    float q1 = __shfl(d1[1], m, 32);
    float q2 = __shfl(d1[2], m, 32);
    float p0 = hi ? q0 : d0[0];
    float p1 = hi ? q1 : d0[1];
    float p2 = hi ? q2 : d0[2];

    float xs = p0 / p2;
    float ys = p1 / p2;

    // --- bilinear sample setup ---
    float x0f = floorf(xs), y0f = floorf(ys);
    int xi = (int)x0f, yi = (int)y0f;
    int xi1 = xi + 1, yi1 = yi + 1;
    float wx = xs - x0f, wy = ys - y0f;
    float vx0 = (xi  >= 0 && xi  <= 255) ? 1.f : 0.f;
    float vx1 = (xi1 >= 0 && xi1 <= 255) ? 1.f : 0.f;
    float vy0 = (yi  >= 0 && yi  <= 255) ? 1.f : 0.f;
    float vy1 = (yi1 >= 0 && yi1 <= 255) ? 1.f : 0.f;
    int xc0 = min(max(xi, 0), 255),  xc1 = min(max(xi1, 0), 255);
    int yc0 = min(max(yi, 0), 255),  yc1 = min(max(yi1, 0), 255);
    float g00 = vx0 * vy0, g01 = vx1 * vy0, g10 = vx0 * vy1, g11 = vx1 * vy1;
    float wA = (1.f - wx) * (1.f - wy);
    float wB = wx * (1.f - wy);
    float wC = (1.f - wx) * wy;
    float wD = wx * wy;
    int o00 = yc0 * WD + xc0, o01 = yc0 * WD + xc1;
    int o10 = yc1 * WD + xc0, o11 = yc1 * WD + xc1;

    // --- light mask (max of sqrt == sqrt of max; monotone) ---
    float lx = (float)light_xy[0], ly = (float)light_xy[1];
    float aa = ab[0], bb = ab[1];
    float lxm = lx - 255.f, lym = ly - 255.f;
    float dA = lx * lx + ly * ly;
    float dB = lxm * lxm + ly * ly;
    float dC = lx * lx + lym * lym;
    float dDc = lxm * lxm + lym * lym;
    float maxlen = sqrtf(fmaxf(fmaxf(dA, dB), fmaxf(dC, dDc)));
    float dr = (float)row - lx;
    float dc = (float)col - ly;
    float L = sqrtf(dr * dr + dc * dc) / maxlen * (aa - bb) + bb;

    const float* base = img + (size_t)b * 3 * HW;
    size_t outBase = (size_t)b * 3 * HW + (size_t)row * WD + col;
    int pixIdx = row * WD + col;

#pragma unroll
    for (int c = 0; c < 3; ++c) {
        const float* ch = base + (size_t)c * HW;
        float v00 = ch[o00] * g00;
        float v01 = ch[o01] * g01;
        float v10 = ch[o10] * g10;
        float v11 = ch[o11] * g11;
        float warped = v00 * wA + v01 * wB + v10 * wC + v11 * wD;
        float mo2 = PRE ? mo_pre[c * HW + pixIdx] : moire2(row, col, c, moire_c, theta);
        size_t oidx = outBase + (size_t)c * HW;
        float nz = noise[oidx];
        out[oidx] = warped * L * 0.85f + mo2 * 0.15f + 0.031622776601683794f * nz;
    }
}

extern "C" void kernel_launch(void* const* d_in, const int* in_sizes, int n_in,
                              void* d_out, int out_size, void* d_ws, size_t ws_size,
                              hipStream_t stream) {
    (void)in_sizes; (void)n_in; (void)out_size;
    const float* img      = (const float*)d_in[0];   // embed_image (B,C,H,W)
    const float* offs     = (const float*)d_in[1];   // dst_offsets (B,4,2)
    const float* ab       = (const float*)d_in[2];   // ab (2,)
    const float* moire_c  = (const float*)d_in[3];   // moire_c (3,2)
    const float* noise    = (const float*)d_in[4];   // noise (B,C,H,W)
    const int*   light_xy = (const int*)d_in[5];     // (2,)
    const int*   theta    = (const int*)d_in[6];     // (3,)
    float* out = (float*)d_out;

    // ws layout: [moire 3*HW floats][hinv BD*9 floats]
    const size_t need = ((size_t)3 * HW + (size_t)BD * 9) * sizeof(float);
    float* mo_ws   = (float*)d_ws;
    float* hinv_ws = (float*)d_ws + (size_t)3 * HW;

    dim3 grid(BD * HD), block(256);
    if (ws_size >= need) {
        hinv_prep_kernel<<<dim3(1), dim3(64), 0, stream>>>(offs, hinv_ws);
        moire_prep_kernel<<<dim3((3 * HW + 255) / 256), block, 0, stream>>>(moire_c, theta, mo_ws);
        shoot_kernel<true><<<grid, block, 0, stream>>>(img, offs, ab, moire_c, noise,
                                                       light_xy, theta, mo_ws, hinv_ws, out);
    } else {
        shoot_kernel<false><<<grid, block, 0, stream>>>(img, offs, ab, moire_c, noise,
                                                        light_xy, theta, nullptr, nullptr, out);
    }
}